// DecorrelatedBN_12266426598076
// MI455X (gfx1250) — compile-verified
//
#include <hip/hip_runtime.h>

typedef __attribute__((ext_vector_type(2))) float v2f;
typedef __attribute__((ext_vector_type(8))) float v8f;

#define GSIZE   16          // channels per group
#define NGROUP  16          // 256 / 16
#define CCH     256
#define EPSV    1e-5f
#define NS_ITERS 14

// ws layout (floats):
//   [0, 256)        per-channel raw sums (mean = sum * inv_n)
//   [256, 4352)     per-group raw X^T X, row-major 16x16 per group
//   [4352, 8448)    proj = (Sigma + eps I)^(-1/2), row-major per group
#define WS_XX   CCH
#define WS_PROJ (CCH + NGROUP * 256)

__global__ void dbn_init(float* ws) {
    int i = blockIdx.x * blockDim.x + threadIdx.x;
    if (i < WS_PROJ) ws[i] = 0.f;
}

// ---------------------------------------------------------------------------
// Phase 1: per-channel sums + per-group X^T X via chained v_wmma_f32_16x16x4_f32.
// A-operand (16x4 fp32): lane L<16 -> {x[r+0][chL], x[r+1][chL]},
//                        lane L>=16 -> {x[r+2][ch(L-16)], x[r+3][ch(L-16)]}
// identical register content to the B-operand (4x16) for the same rows, so one
// register pair feeds both sides of X^T X. 16 rows (4 quads) per iteration:
// 8 loads in flight, 4 chained wmmas. Uniform scalar trip count.
// ---------------------------------------------------------------------------
__global__ __launch_bounds__(256) void dbn_stats(const float* __restrict__ x,
                                                 float* __restrict__ ws,
                                                 int iters) {
    int lane = threadIdx.x & 31;
    int wid  = threadIdx.x >> 5;
    int gw   = blockIdx.x * 8 + wid;
    int g    = gw & (NGROUP - 1);
    int slice = gw >> 4;

    int half = lane >> 4;            // 0: quad rows k=0,1   1: quad rows k=2,3
    int ch   = lane & 15;
    int cidx = g * GSIZE + ch;

    // element offset of this lane's first value (rows advance 16 per iter)
    unsigned off = (unsigned)(slice * iters * 16 + half * 2) * CCH + (unsigned)cidx;

    v8f c = {};
    float s = 0.f;
    for (int it = 0; it < iters; ++it) {          // uniform -> scalar loop
        float v0 = x[off + 0 * CCH];
        float v1 = x[off + 1 * CCH];
        float v2 = x[off + 4 * CCH];
        float v3 = x[off + 5 * CCH];
        float v4 = x[off + 8 * CCH];
        float v5 = x[off + 9 * CCH];
        float v6 = x[off + 12 * CCH];
        float v7 = x[off + 13 * CCH];
        off += 16u * CCH;
        v2f q0 = { v0, v1 };
        v2f q1 = { v2, v3 };
        v2f q2 = { v4, v5 };
        v2f q3 = { v6, v7 };
        c = __builtin_amdgcn_wmma_f32_16x16x4_f32(false, q0, false, q0, (short)0, c, false, false);
        c = __builtin_amdgcn_wmma_f32_16x16x4_f32(false, q1, false, q1, (short)0, c, false, false);
        c = __builtin_amdgcn_wmma_f32_16x16x4_f32(false, q2, false, q2, (short)0, c, false, false);
        c = __builtin_amdgcn_wmma_f32_16x16x4_f32(false, q3, false, q3, (short)0, c, false, false);
        s += (v0 + v1) + (v2 + v3) + ((v4 + v5) + (v6 + v7));
    }
    atomicAdd(&ws[cidx], s);
    float* xx = ws + WS_XX + g * 256;
    #pragma unroll
    for (int r = 0; r < 8; ++r) {
        int i = r + half * 8;        // C/D layout: vgpr r -> M = r (+8 for hi lanes)
        atomicAdd(&xx[i * 16 + ch], c[r]);
    }
}

// ---------------------------------------------------------------------------
// Phase 2: sigma = XX/n - mean mean^T + eps I ; proj = sigma^(-1/2) via
// coupled Newton-Schulz:  Y0=A, Z0=I, W=0.5(3I - Z Y), Y<-YW, Z<-WZ,
// Z -> A^(-1/2); un-normalize by 1/sqrt(trace). One block per group.
// ---------------------------------------------------------------------------
__global__ __launch_bounds__(256) void dbn_newton(float* __restrict__ ws, float inv_n) {
    __shared__ float Ys[256], Zs[256], Wsh[256];
    int g = blockIdx.x;
    int tid = threadIdx.x;
    int i = tid >> 4, j = tid & 15;

    const float* xx = ws + WS_XX + g * 256;
    float mi = ws[g * 16 + i] * inv_n;
    float mj = ws[g * 16 + j] * inv_n;
    float sig = xx[tid] * inv_n - mi * mj + (i == j ? EPSV : 0.f);
    Wsh[tid] = sig;
    __syncthreads();
    float tr = 0.f;
    #pragma unroll
    for (int k = 0; k < 16; ++k) tr += Wsh[k * 16 + k];
    float a = sig / tr;
    Ys[tid] = a;
    Zs[tid] = (i == j) ? 1.f : 0.f;
    for (int it = 0; it < NS_ITERS; ++it) {
        __syncthreads();
        float t = 0.f;
        #pragma unroll
        for (int k = 0; k < 16; ++k) t += Zs[i * 16 + k] * Ys[k * 16 + j];
        float w = 0.5f * ((i == j ? 3.f : 0.f) - t);
        Wsh[tid] = w;
        __syncthreads();
        float yn = 0.f, zn = 0.f;
        #pragma unroll
        for (int k = 0; k < 16; ++k) {
            yn += Ys[i * 16 + k] * Wsh[k * 16 + j];
            zn += Wsh[i * 16 + k] * Zs[k * 16 + j];
        }
        __syncthreads();
        Ys[tid] = yn;
        Zs[tid] = zn;
    }
    __syncthreads();
    ws[WS_PROJ + g * 256 + tid] = Zs[tid] * rsqrtf(tr);
}

// ---------------------------------------------------------------------------
// Phase 3: out = (x - mean) @ proj (per group), * gamma + beta.
// Each wave: one group, proj kept in B-layout regs; 2 row-tiles per iteration
// (8 b64 loads in flight, 8 wmmas), nontemporal output stores (write-only data;
// keeps x resident in L2 for this pass's reads).
// ---------------------------------------------------------------------------
__global__ __launch_bounds__(256) void dbn_apply(const float* __restrict__ x,
                                                 const float* __restrict__ gamma,
                                                 const float* __restrict__ beta,
                                                 const float* __restrict__ ws,
                                                 float* __restrict__ out,
                                                 int wave_stride, int tiles_per_wave,
                                                 float inv_n) {
    int lane = threadIdx.x & 31;
    int wid  = threadIdx.x >> 5;
    int gw   = blockIdx.x * 8 + wid;
    int g    = gw & (NGROUP - 1);
    int tbase = gw >> 4;

    int half = lane >> 4;
    int ch   = lane & 15;

    // proj in B layout: slice s covers K rows s*4 .. s*4+3
    const float* proj = ws + WS_PROJ + g * 256;
    v2f b[4];
    #pragma unroll
    for (int s2 = 0; s2 < 4; ++s2) {
        int i0 = s2 * 4 + half * 2;
        b[s2].x = proj[i0 * 16 + ch];
        b[s2].y = proj[(i0 + 1) * 16 + ch];
    }
    // per-lane means matching the A-operand channel pairs
    v2f mn[4];
    #pragma unroll
    for (int s2 = 0; s2 < 4; ++s2) {
        int c0 = g * GSIZE + s2 * 4 + half * 2;
        mn[s2].x = ws[c0] * inv_n;
        mn[s2].y = ws[c0 + 1] * inv_n;
    }
    float gm = gamma[g * GSIZE + ch];
    float bt = beta[g * GSIZE + ch];

    for (int t = 0; t < tiles_per_wave; t += 2) {
        unsigned tile0 = (unsigned)(tbase + t * wave_stride);
        unsigned tile1 = (unsigned)(tbase + (t + 1) * wave_stride);
        unsigned inA0 = (tile0 * 16 + ch) * CCH + g * GSIZE + half * 2;
        unsigned inA1 = (tile1 * 16 + ch) * CCH + g * GSIZE + half * 2;

        v2f a0[4], a1[4];
        #pragma unroll
        for (int s2 = 0; s2 < 4; ++s2) a0[s2] = *(const v2f*)(x + inA0 + s2 * 4);
        #pragma unroll
        for (int s2 = 0; s2 < 4; ++s2) a1[s2] = *(const v2f*)(x + inA1 + s2 * 4);

        v8f c0 = {}, c1 = {};
        #pragma unroll
        for (int s2 = 0; s2 < 4; ++s2) {
            v2f a = a0[s2];
            a.x -= mn[s2].x; a.y -= mn[s2].y;
            c0 = __builtin_amdgcn_wmma_f32_16x16x4_f32(false, a, false, b[s2], (short)0, c0, false, false);
        }
        #pragma unroll
        for (int s2 = 0; s2 < 4; ++s2) {
            v2f a = a1[s2];
            a.x -= mn[s2].x; a.y -= mn[s2].y;
            c1 = __builtin_amdgcn_wmma_f32_16x16x4_f32(false, a, false, b[s2], (short)0, c1, false, false);
        }

        unsigned ob0 = (tile0 * 16 + half * 8) * CCH + g * GSIZE + ch;
        unsigned ob1 = (tile1 * 16 + half * 8) * CCH + g * GSIZE + ch;
        #pragma unroll
        for (int r = 0; r < 8; ++r)
            __builtin_nontemporal_store(c0[r] * gm + bt, out + ob0 + r * CCH);
        #pragma unroll
        for (int r = 0; r < 8; ++r)
            __builtin_nontemporal_store(c1[r] * gm + bt, out + ob1 + r * CCH);
    }
}

extern "C" void kernel_launch(void* const* d_in, const int* in_sizes, int n_in,
                              void* d_out, int out_size, void* d_ws, size_t ws_size,
                              hipStream_t stream) {
    const float* x     = (const float*)d_in[0];
    const float* gamma = (const float*)d_in[1];
    const float* beta  = (const float*)d_in[2];
    float* out = (float*)d_out;
    float* ws  = (float*)d_ws;

    int nrows = in_sizes[0] / CCH;            // 128*32*32 = 131072
    float inv_n = 1.0f / (float)nrows;

    // zero the accumulators (deterministic per call)
    dbn_init<<<(WS_PROJ + 255) / 256, 256, 0, stream>>>(ws);

    // phase 1: stats. 256 waves per group, 16 rows per iteration
    int wpg = 256;
    int iters = nrows / (wpg * 16);           // 32
    dbn_stats<<<NGROUP * wpg / 8, 256, 0, stream>>>(x, ws, iters);

    // phase 2: Newton-Schulz inverse sqrt, one block per group
    dbn_newton<<<NGROUP, 256, 0, stream>>>(ws, inv_n);

    // phase 3: apply. 512 waves per group, 16-row tiles, 2 tiles/iter
    int tiles = nrows / 16;                   // 8192
    int wave_stride = 512;
    int tpw = tiles / wave_stride;            // 16
    dbn_apply<<<NGROUP * wave_stride / 8, 256, 0, stream>>>(
        x, gamma, beta, ws, out, wave_stride, tpw, inv_n);
}